// PointNet_18279380812435
// MI455X (gfx1250) — compile-verified
//
#include <hip/hip_runtime.h>

typedef __attribute__((ext_vector_type(2))) float v2f;
typedef __attribute__((ext_vector_type(8))) float v8f;

#define NODES 100000
#define KNN   6
#define NPW   8              // nodes per workgroup
#define EPW   (NPW * KNN)    // 48 edges per workgroup
#define MT    (EPW / 16)     // 3 M-tiles of 16 edges
#define WAVES 8              // 256 threads

// GEMM: Out[EPW][NDIM] (+bias, opt relu)  =  As[EPW][KDIM] @ W[KDIM][NDIM]
// Tile t = wave + i*8. Since NT | 8, t % NT == wave % NT: each wave keeps one
// N-tile (B operand CSEs across the unroll) and sweeps M-tiles
// mt_i = wave/NT + i*(8/NT), wrapped mod MT with a branch-free scalar select.
// Waves whose slots wrap recompute a tile another wave also computes —
// identical values to identical LDS addresses, benign — so every wave runs an
// identical, branch-free, fully unrolled instruction stream (EXEC all-ones
// around every WMMA, per ISA §7.12).
// KCLAMP: highest valid weight row (+1); A is zero-padded above it.
template <int KDIM, int KCLAMP, int NDIM, bool RELU>
__device__ __forceinline__ void gemm_tiles(
    const float* __restrict__ As,    // LDS, row stride KDIM
    const float* __restrict__ W,     // global, row-major
    const float* __restrict__ bias,  // [NDIM]
    float* __restrict__ Out,         // LDS, row stride NDIM
    int lane, int wave)              // wave: SGPR-uniform (readfirstlane)
{
    constexpr int NT  = NDIM / 16;                    // n-tiles (2/4/8)
    constexpr int WPN = WAVES / NT;                   // waves per n-tile
    constexpr int TPW = (MT * NT + WAVES - 1) / WAVES; // tiles per wave (const)

    const int m   = lane & 15;       // A row / B,D column inside tile
    const int kh  = (lane >> 4) * 2; // K pair selected by lane half
    const int nt  = wave % NT;       // scalar, invariant across i
    const int mt0 = wave / NT;       // scalar
    const int c0  = nt * 16;         // scalar

    v8f acc[TPW];
    #pragma unroll
    for (int i = 0; i < TPW; ++i) acc[i] = {};

    #pragma unroll
    for (int k0 = 0; k0 < KDIM; k0 += 4) {
        v2f b;
        int k1 = k0 + kh;     k1 = (k1 < KCLAMP) ? k1 : (KCLAMP - 1);
        int k2 = k0 + kh + 1; k2 = (k2 < KCLAMP) ? k2 : (KCLAMP - 1);
        b.x = W[k1 * NDIM + c0 + m];
        b.y = W[k2 * NDIM + c0 + m];
        #pragma unroll
        for (int i = 0; i < TPW; ++i) {
            int mt = mt0 + i * WPN;
            mt = (mt >= MT) ? (mt - MT) : mt;          // scalar select, no branch
            const v2f a = *(const v2f*)&As[(mt * 16 + m) * KDIM + k0 + kh];
            acc[i] = __builtin_amdgcn_wmma_f32_16x16x4_f32(
                false, a, false, b, (short)0, acc[i], false, false);
        }
    }

    const float bs = bias[c0 + m];
    #pragma unroll
    for (int i = 0; i < TPW; ++i) {
        int mt = mt0 + i * WPN;
        mt = (mt >= MT) ? (mt - MT) : mt;
        const int rowbase = mt * 16 + (lane >> 4) * 8;  // D: lanes16-31 -> M+8
        #pragma unroll
        for (int v = 0; v < 8; ++v) {
            float x = acc[i][v] + bs;
            if (RELU) x = (x > 0.0f) ? x : 0.0f;
            Out[(rowbase + v) * NDIM + (c0 + m)] = x;
        }
    }
}

// One fused PointNetConv layer:
//   m[e]   = relu(concat(h[src], pos[src]-pos[dst]) @ Wa + ba) @ Wb + bb
//   out[i] = relu(max_{e in edges(i)} m[e])
template <int CIN, int H, int COUT>
__global__ __launch_bounds__(256) void pn_layer(
    const float* __restrict__ h_in,   // [N, CIN]
    const float* __restrict__ pos,    // [N, 3]
    const int*   __restrict__ ei,     // [2, E] row0=src row1=dst
    const float* __restrict__ Wa,     // [CIN+3, H]
    const float* __restrict__ ba,     // [H]
    const float* __restrict__ Wb,     // [H, COUT]
    const float* __restrict__ bb,     // [COUT]
    float* __restrict__ h_out)        // [N, COUT]
{
    constexpr int FEAT = CIN + 3;
    constexpr int KP   = (FEAT + 3) & ~3;              // K padded to mult of 4
    constexpr int FMW  = (KP > COUT) ? KP : COUT;      // F / M2 share one buffer

    __shared__ __align__(16) float Ts[EPW * H];        // hidden activations
    __shared__ __align__(16) float FM[EPW * FMW];      // features, then messages
    __shared__ int s_src[EPW];
    __shared__ int s_dst[EPW];

    const int tid  = threadIdx.x;
    const int lane = tid & 31;
    // SGPR-uniform wave id: all scheduling math is scalar, EXEC never changes.
    const int wave = __builtin_amdgcn_readfirstlane(tid >> 5);
    const int n0   = blockIdx.x * NPW;                 // first node of this WG
    const int e0   = n0 * KNN;                         // first edge of this WG
    const int E    = NODES * KNN;

    if (tid < EPW) {
        s_src[tid] = ei[e0 + tid];
        s_dst[tid] = ei[E + e0 + tid];
    }
    __syncthreads();

    // ---- Phase 1: gather edge features into FM (row stride KP, zero-padded K)
    for (int idx = tid; idx < EPW * KP; idx += 256) {
        int e = idx / KP;
        int c = idx - e * KP;
        float v = 0.0f;
        int s = s_src[e];
        if (c < CIN) {
            v = h_in[s * CIN + c];
        } else if (c < FEAT) {
            int cc = c - CIN;
            v = pos[s * 3 + cc] - pos[s_dst[e] * 3 + cc];
        }
        FM[idx] = v;
    }
    __syncthreads();

    // ---- Phase 2: Ts = relu(F @ Wa + ba)
    gemm_tiles<KP, FEAT, H, true>(FM, Wa, ba, Ts, lane, wave);
    __syncthreads();

    // ---- Phase 3: M2 = Ts @ Wb + bb  (into FM; F is dead)
    gemm_tiles<H, H, COUT, false>(Ts, Wb, bb, FM, lane, wave);
    __syncthreads();

    // ---- Phase 4: out = relu(max over the 6 contiguous edges per node)
    for (int idx = tid; idx < NPW * COUT; idx += 256) {
        const int nl = idx / COUT;
        const int c  = idx - nl * COUT;
        const float* p = &FM[nl * KNN * COUT + c];
        float mx = 0.0f;                     // fused outer relu
        #pragma unroll
        for (int j = 0; j < KNN; ++j) {
            float x = p[j * COUT];
            mx = (x > mx) ? x : mx;
        }
        h_out[(n0 + nl) * COUT + c] = mx;
    }
}

extern "C" void kernel_launch(void* const* d_in, const int* in_sizes, int n_in,
                              void* d_out, int out_size, void* d_ws, size_t ws_size,
                              hipStream_t stream) {
    (void)in_sizes; (void)n_in; (void)out_size; (void)ws_size;
    const float* pos = (const float*)d_in[0];
    const int*   ei  = (const int*)d_in[1];
    const float* W1a = (const float*)d_in[2];
    const float* b1a = (const float*)d_in[3];
    const float* W1b = (const float*)d_in[4];
    const float* b1b = (const float*)d_in[5];
    const float* W2a = (const float*)d_in[6];
    const float* b2a = (const float*)d_in[7];
    const float* W2b = (const float*)d_in[8];
    const float* b2b = (const float*)d_in[9];
    const float* W3a = (const float*)d_in[10];
    const float* b3a = (const float*)d_in[11];
    const float* W3b = (const float*)d_in[12];
    const float* b3b = (const float*)d_in[13];

    float* h1 = (float*)d_ws;                 // [N, 32]  = 12.8 MB
    float* h2 = h1 + (size_t)NODES * 32;      // [N, 64]  = 25.6 MB
    float* h3 = (float*)d_out;                // [N, 128]

    dim3 grid(NODES / NPW), block(256);
    pn_layer<3, 32, 32><<<grid, block, 0, stream>>>(pos, pos, ei, W1a, b1a, W1b, b1b, h1);
    pn_layer<32, 64, 64><<<grid, block, 0, stream>>>(h1, pos, ei, W2a, b2a, W2b, b2b, h2);
    pn_layer<64, 128, 128><<<grid, block, 0, stream>>>(h2, pos, ei, W3a, b3a, W3b, b3b, h3);
}